// HungarianMatcher_50276887167550
// MI455X (gfx1250) — compile-verified
//
#include <hip/hip_runtime.h>

typedef __attribute__((ext_vector_type(2))) float v2f;
typedef __attribute__((ext_vector_type(8))) float v8f;

#define NB 64
#define NN 128
#define NT 100

// ---------------------------------------------------------------------------
// Kernel A: build Cp[b][m][n] = 10*softplus(-logit_m) + |px_m-tx_n| + |py_m-ty_n|
// (zero for n >= NT) using V_WMMA_F32_16X16X4_F32 as a fused rank-1 outer
// product + accumulate:  D = (10*cc) x ones^T + L1.
// One wave32 per 16x16 tile; grid = B * 8 * 8.
// ---------------------------------------------------------------------------
__global__ __launch_bounds__(32)
void cost_tiles_wmma(const float* __restrict__ outs,   // [B,128,3]
                     const float* __restrict__ tgts,   // [B,100,3]
                     float* __restrict__ cost) {       // [B,128,128]
  const int bid  = blockIdx.x;
  const int b    = bid >> 6;
  const int tm   = (bid >> 3) & 7;
  const int tn   = bid & 7;
  const int lane = threadIdx.x;
  const int half = lane >> 4;   // 0: lanes 0-15, 1: lanes 16-31
  const int l16  = lane & 15;
  const int rb   = tm * 16;
  const int cb   = tn * 16;

  // ---- A fragment (16x4 f32): only K=0 slot nonzero = 10*class_cost(row) ---
  const int mA = rb + l16;
  const float x  = outs[(b * NN + mA) * 3 + 0];
  // -log_sigmoid(x) = softplus(-x) = max(-x,0) + log1p(exp(-|x|))
  const float cc = 10.0f * (fmaxf(-x, 0.0f) + log1pf(expf(-fabsf(x))));
  v2f a;
  a.x = (half == 0) ? cc : 0.0f;
  a.y = 0.0f;

  // ---- B fragment (4x16 f32): ones in the K=0 row ----
  v2f bm;
  bm.x = (half == 0) ? 1.0f : 0.0f;
  bm.y = 0.0f;

  // ---- C fragment = L1 cost tile; column of this lane is nn = cb + l16 ----
  const int  nn    = cb + l16;
  const int  nc    = (nn < NT) ? nn : (NT - 1);   // clamp to avoid divergence
  const bool valid = (nn < NT);
  const float tx = tgts[(b * NT + nc) * 3 + 1];
  const float ty = tgts[(b * NT + nc) * 3 + 2];

  v8f c;
#pragma unroll
  for (int r = 0; r < 8; ++r) {
    const int m  = rb + r + 8 * half;             // C/D: row = vgpr + 8*half
    const float px = outs[(b * NN + m) * 3 + 1];
    const float py = outs[(b * NN + m) * 3 + 2];
    c[r] = valid ? (fabsf(px - tx) + fabsf(py - ty)) : 0.0f;
  }

  // D = A x B + C   (v_wmma_f32_16x16x4_f32)
  v8f d = __builtin_amdgcn_wmma_f32_16x16x4_f32(
      /*neg_a=*/false, a, /*neg_b=*/false, bm,
      /*c_mod=*/(short)0, c, /*reuse_a=*/false, /*reuse_b=*/false);

#pragma unroll
  for (int r = 0; r < 8; ++r) {
    const int m = rb + r + 8 * half;
    cost[(size_t)b * NN * NN + (size_t)m * NN + nn] = valid ? d[r] : 0.0f;
  }
}

// ---------------------------------------------------------------------------
// Kernel B: shortest-augmenting-path Hungarian, one block (128 threads = 4
// wave32) per batch. The 64KB cost matrix is staged into LDS once via the
// CDNA5 async global->LDS path (GLOBAL_LOAD_ASYNC_TO_LDS_B128 + ASYNCcnt),
// so every serial inner step reads its cost row from LDS instead of L2.
// Argmin per step: __shfl_xor butterfly inside each wave (exact, keeps the
// jnp.argmin lowest-index tie-break) + 4-way cross-wave combine -> only 3
// barriers per serial step.
// ---------------------------------------------------------------------------
__global__ __launch_bounds__(128)
void hungarian_kernel(const float* __restrict__ cost,   // [B,128,128]
                      long long* __restrict__ pred_idx, // [B,100]
                      long long* __restrict__ tgt_idx) {// [B,100]
  const int b = blockIdx.x;
  const int t = threadIdx.x;                 // column j = t + 1
  const int w = t >> 5;                      // wave id 0..3
  const int lane = t & 31;
  const float* __restrict__ Cb = cost + (size_t)b * NN * NN;

  __shared__ float lds_cost[NN * NN];        // 64 KB
  __shared__ float u[NN + 1], v[NN + 1], minv[NN + 1];
  __shared__ int   p[NN + 1], way[NN + 1], usedA[NN + 1];
  __shared__ float red_val[4];
  __shared__ int   red_idx[4];
  __shared__ int   sh_j0;

  const float INF = 1000000000.0f;

  // ---- async-stage the batch cost matrix into LDS (4096 x B128) ----
  {
    const unsigned int lbase = (unsigned int)(uintptr_t)(&lds_cost[0]);
#pragma unroll 4
    for (int k = 0; k < 32; ++k) {
      const unsigned int byteoff = (unsigned int)(k * NN + t) * 16u;
      const unsigned int loff = lbase + byteoff;
      const unsigned long long gaddr =
          (unsigned long long)(uintptr_t)((const char*)Cb + byteoff);
      asm volatile("global_load_async_to_lds_b128 %0, %1, off"
                   :: "v"(loff), "v"(gaddr)
                   : "memory");
    }
    asm volatile("s_wait_asynccnt 0" ::: "memory");
  }

  u[t] = 0.0f; v[t] = 0.0f; p[t] = 0;
  if (t == 0) { u[NN] = 0.0f; v[NN] = 0.0f; p[NN] = 0; }
  __syncthreads();

  for (int i = 1; i <= NN; ++i) {
    if (t == 0) { p[0] = i; sh_j0 = 0; usedA[0] = 0; }
    minv[t + 1] = INF; way[t + 1] = 0; usedA[t + 1] = 0;
    __syncthreads();

    // ---- augmenting path search ----
    while (true) {
      const int j0 = sh_j0;                  // uniform across block
      if (p[j0] == 0) break;                 // found a free column
      if (t == 0) usedA[j0] = 1;
      __syncthreads();

      const int   i0  = p[j0];
      const float ui0 = u[i0];
      const int   j   = t + 1;
      const int   isused = usedA[j];
      if (!isused) {
        const float cur = lds_cost[(i0 - 1) * NN + t] - ui0 - v[j];
        if (cur < minv[j]) { minv[j] = cur; way[j] = j0; }
      }

      // ---- argmin (value, lowest index) ----
      float rv = isused ? INF : minv[j];
      int   ri = j;
#pragma unroll
      for (int s = 1; s < 32; s <<= 1) {
        const float ov = __shfl_xor(rv, s, 32);
        const int   oi = __shfl_xor(ri, s, 32);
        if (ov < rv || (ov == rv && oi < ri)) { rv = ov; ri = oi; }
      }
      if (lane == 0) { red_val[w] = rv; red_idx[w] = ri; }
      __syncthreads();
      float delta = red_val[0];
      int   j1    = red_idx[0];
#pragma unroll
      for (int ww = 1; ww < 4; ++ww) {        // ascending idx order: exact
        const float ov = red_val[ww];
        const int   oi = red_idx[ww];
        if (ov < delta || (ov == delta && oi < j1)) { delta = ov; j1 = oi; }
      }

      // potential updates (scatter targets p[j] are distinct for used j)
      if (isused) { u[p[j]] += delta; v[j] -= delta; }
      else        { minv[j] -= delta; }
      if (t == 0) {
        if (usedA[0]) u[p[0]] += delta;       // j=0 contribution: u[i] += delta
        sh_j0 = j1;
      }
      __syncthreads();
    }

    // ---- flip matching along the augmenting path (serial, thread 0) ----
    if (t == 0) {
      int j0 = sh_j0;
      while (j0 != 0) {
        const int j1 = way[j0];
        p[j0] = p[j1];
        j0 = j1;
      }
    }
    __syncthreads();
  }

  if (t < NT) {
    pred_idx[b * NT + t] = (long long)(p[t + 1] - 1);
    tgt_idx [b * NT + t] = (long long)t;
  }
}

extern "C" void kernel_launch(void* const* d_in, const int* in_sizes, int n_in,
                              void* d_out, int out_size, void* d_ws, size_t ws_size,
                              hipStream_t stream) {
  (void)in_sizes; (void)n_in; (void)out_size; (void)ws_size;
  const float* outs = (const float*)d_in[0];   // [64,128,3] f32
  const float* tgts = (const float*)d_in[1];   // [64,100,3] f32
  float* cost = (float*)d_ws;                  // 64*128*128*4 = 4 MB scratch
  long long* out = (long long*)d_out;          // [pred_idx | tgt_idx], int64

  cost_tiles_wmma<<<dim3(NB * 64), dim3(32), 0, stream>>>(outs, tgts, cost);
  hungarian_kernel<<<dim3(NB), dim3(128), 0, stream>>>(cost, out, out + NB * NT);
}